// SumAggregator_46677704573422
// MI455X (gfx1250) — compile-verified
//
#include <hip/hip_runtime.h>
#include <stdint.h>

// SumAggregator: out[n, :] = sum_{j<32} emb[neighs[n*32+j], :]
// DIM=128 fp32 -> one neighbor row = 512B = one wave32-wide b128 transfer.
// One node per wave; async global->LDS DMA ring (ASYNCcnt) for deep MLP with
// minimal VGPR cost; indices preloaded into SGPRs (single kmcnt wait);
// GVS-mode addressing (saddr = table base, 32-bit per-lane voff).

#define DIM 128
#define NB 32
#define WAVES_PER_BLOCK 8
#define SLOTS 8                      // in-flight neighbor rows per wave
#define ROW_BYTES (DIM * 4)          // 512 B
// LDS usage: 8 waves * 8 slots * 512B = 32 KB per workgroup

typedef float float4v __attribute__((ext_vector_type(4)));

// Issue one async row-load in GVS mode: saddr = emb base (SGPR pair),
// voff = idx*512 + lane*16 (32-bit VGPR). s_wait_dscnt 0 first so the DMA
// write cannot overwrite a slot whose previous ds_load is still in flight.
__device__ __forceinline__ void async_row_b128(uint32_t lds_off, uint32_t voff,
                                               uint64_t sbase) {
    asm volatile("s_wait_dscnt 0\n\t"
                 "global_load_async_to_lds_b128 %0, %1, %2"
                 :
                 : "v"(lds_off), "v"(voff), "s"(sbase)
                 : "memory");
}

__device__ __forceinline__ void wait_async_le_slots_m1() {
    asm volatile("s_wait_asynccnt 7" ::: "memory");   // SLOTS-1
}
__device__ __forceinline__ void wait_async_zero() {
    asm volatile("s_wait_asynccnt 0" ::: "memory");
}

__global__ __launch_bounds__(WAVES_PER_BLOCK * 32)
void SumAggregator_46677704573422_kernel(const int* __restrict__ neighs,
                                         const float* __restrict__ emb,
                                         float* __restrict__ out,
                                         int node_count) {
    __shared__ float lds[WAVES_PER_BLOCK * SLOTS * DIM];

    const int lane = threadIdx.x & 31;
    const int wave = threadIdx.x >> 5;
    // node is wave-uniform; force into SGPR so index loads become s_load.
    const int node =
        __builtin_amdgcn_readfirstlane((int)(blockIdx.x * WAVES_PER_BLOCK + wave));
    if (node >= node_count) return;

    // Flat pointers to __shared__ carry the LDS byte offset in their low 32
    // bits (LDS aperture).
    float* wave_slots = &lds[(wave * SLOTS) * DIM];
    const uint32_t lds_wave_base = (uint32_t)(uintptr_t)wave_slots;
    const uint32_t lane_b = (uint32_t)(lane * 16);
    const uint64_t emb_base = (uint64_t)(uintptr_t)emb;

    // ---- preload all 32 neighbor indices (uniform -> wide s_load, 1 wait) --
    const int* __restrict__ nptr = neighs + (size_t)node * NB;
    uint32_t idx[NB];
#pragma unroll
    for (int j = 0; j < NB; ++j) idx[j] = (uint32_t)nptr[j];

    // ---- prologue: fill the ring with SLOTS async row loads ----
#pragma unroll
    for (int j = 0; j < SLOTS; ++j) {
        async_row_b128(lds_wave_base + (uint32_t)j * ROW_BYTES + lane_b,
                       (idx[j] << 9) + lane_b, emb_base);
    }

    float4v acc = {0.0f, 0.0f, 0.0f, 0.0f};

    // ---- steady state: consume oldest slot, refill it ----
#pragma unroll
    for (int j = 0; j < NB - SLOTS; ++j) {
        wait_async_le_slots_m1();                 // oldest of 8 in-flight done
        const int slot = j & (SLOTS - 1);
        const float4v v =
            *(const float4v*)&wave_slots[slot * DIM + lane * 4]; // ds_load_b128
        acc += v;
        async_row_b128(lds_wave_base + (uint32_t)slot * ROW_BYTES + lane_b,
                       (idx[j + SLOTS] << 9) + lane_b, emb_base);
    }

    // ---- drain: all issued; wait for everything, consume remaining slots ----
    wait_async_zero();
#pragma unroll
    for (int j = NB - SLOTS; j < NB; ++j) {
        const int slot = j & (SLOTS - 1);
        const float4v v = *(const float4v*)&wave_slots[slot * DIM + lane * 4];
        acc += v;
    }

    // coalesced b128 store of this node's 128-dim result
    *(float4v*)(out + (size_t)node * DIM + lane * 4) = acc;
}

extern "C" void kernel_launch(void* const* d_in, const int* in_sizes, int n_in,
                              void* d_out, int out_size, void* d_ws, size_t ws_size,
                              hipStream_t stream) {
    const int* neighs = (const int*)d_in[0];
    // d_in[1] is the node_count scalar on device; graph capture forbids a sync
    // copy, so derive it from the index array length instead.
    const float* emb = (const float*)d_in[2];
    float* out = (float*)d_out;

    const int node_count = in_sizes[0] / NB;
    const int blocks = (node_count + WAVES_PER_BLOCK - 1) / WAVES_PER_BLOCK;

    SumAggregator_46677704573422_kernel<<<blocks, WAVES_PER_BLOCK * 32, 0, stream>>>(
        neighs, emb, out, node_count);
}